// QuadrupletLoss_52733608460725
// MI455X (gfx1250) — compile-verified
//
#include <hip/hip_runtime.h>

// QuadrupletLoss for MI455X (gfx1250, wave32).
// sq -> WMMA f32 Gram + distance matrix -> thr -> triplet (LDS-staged Dm) ->
// quad (compacted positive-pair list in LDS) -> deterministic fixed-order combine.
// Matrix op: V_WMMA_F32_16X16X4_F32 (32 chained issues per 16x16 tile, K=128).

typedef __attribute__((ext_vector_type(2))) float v2f;
typedef __attribute__((ext_vector_type(8))) float v8f;

#define BN 96
#define DD 128
#define NTILE 6                 // 96/16
#define NBLK ((BN * BN) / 256)  // 36
#define NPMAX (BN * (BN - 1))   // worst-case positive pairs (all labels equal)

// ---------------- sq[i] = sum_d E[i][d]^2 -----------------------------------
__global__ void k_sq(const float* __restrict__ E, float* __restrict__ SQ) {
    int i = blockIdx.x * blockDim.x + threadIdx.x;
    if (i < BN) {
        float s = 0.0f;
#pragma unroll 8
        for (int d = 0; d < DD; ++d) { float e = E[i * DD + d]; s += e * e; }
        SQ[i] = s;
    }
}

// ---------------- Gram tile via V_WMMA_F32_16X16X4_F32, then Dm -------------
// One wave per 16x16 tile; EXEC all-ones (WMMA requirement).
// A (16x4): lane L -> row M=L&15; VGPR0/1 = K{0,1} (lanes 0-15) / K{2,3} (16-31).
// B (4x16): B[k][n] = E[tjb+n][k] -> same addressing with column-tile row base.
// C/D: VGPR v, lane L -> M = v + 8*(L>=16), N = L&15.
__global__ void __launch_bounds__(32)
k_gram_dm(const float* __restrict__ E, const float* __restrict__ SQ,
          float* __restrict__ DM) {
    int lane = threadIdx.x;
    int tib = (blockIdx.x / NTILE) * 16;
    int tjb = (blockIdx.x % NTILE) * 16;
    int mn   = lane & 15;
    int koff = (lane >> 4) << 1;
    const float* arow = E + (tib + mn) * DD + koff;
    const float* brow = E + (tjb + mn) * DD + koff;

    v8f c = {};
    for (int k0 = 0; k0 < DD; k0 += 4) {
        v2f a = *(const v2f*)(arow + k0);
        v2f b = *(const v2f*)(brow + k0);
        c = __builtin_amdgcn_wmma_f32_16x16x4_f32(
                false, a, false, b, (short)0, c, false, false);
    }

    int j = tjb + mn;
    float sqj = SQ[j];
    int mbase = (lane >> 4) << 3;
#pragma unroll
    for (int v = 0; v < 8; ++v) {
        int i = tib + mbase + v;
        float d2 = SQ[i] + sqj - 2.0f * c[v];
        DM[i * BN + j] = sqrtf(fmaxf(d2, 0.0f) + 1e-16f);
    }
}

// ---------------- masked means of Dm -> thr = max(mu_neg - mu_pos, 0) -------
__global__ void __launch_bounds__(256)
k_stats(const float* __restrict__ DM, const int* __restrict__ lbl,
        float* __restrict__ thr) {
    __shared__ float sps[256], sns[256];
    __shared__ unsigned spc[256], snc[256];
    __shared__ int sl[BN];
    int t = threadIdx.x;
    if (t < BN) sl[t] = lbl[t];
    __syncthreads();
    float ps = 0.0f, ns = 0.0f; unsigned pc = 0, nc = 0;
    for (int e = t; e < BN * BN; e += 256) {
        int i = e / BN, j = e % BN;
        if (i == j) continue;
        float d = DM[e];
        if (sl[i] == sl[j]) { ps += d; ++pc; } else { ns += d; ++nc; }
    }
    sps[t] = ps; sns[t] = ns; spc[t] = pc; snc[t] = nc;
    __syncthreads();
    for (int s = 128; s > 0; s >>= 1) {
        if (t < s) { sps[t] += sps[t+s]; sns[t] += sns[t+s];
                     spc[t] += spc[t+s]; snc[t] += snc[t+s]; }
        __syncthreads();
    }
    if (t == 0) {
        float mu = sns[0] / (float)snc[0] - sps[0] / (float)spc[0];
        *thr = fmaxf(mu, 0.0f);
    }
}

// ---------------- triplet: per-block partials into fixed slots --------------
__global__ void __launch_bounds__(256)
k_triplet(const float* __restrict__ DM, const int* __restrict__ lbl,
          const float* __restrict__ thrp,
          double* __restrict__ tpart, unsigned* __restrict__ tcnt) {
    __shared__ float sdm[BN * BN];            // 36 KB staged distance matrix
    __shared__ int sl[BN];
    __shared__ double ss[256];
    __shared__ unsigned sc[256];
    int t = threadIdx.x;
    for (int e = t; e < BN * BN; e += 256) sdm[e] = DM[e];
    if (t < BN) sl[t] = lbl[t];
    __syncthreads();
    float thr = *thrp;
    int gid = blockIdx.x * 256 + t;           // exactly BN*BN threads total
    int i = gid / BN, j = gid % BN;
    double sum = 0.0; unsigned cnt = 0;
    if (i != j && sl[i] == sl[j]) {           // positive (i,j) pair
        int li = sl[i];
        float dij = sdm[i * BN + j];
        for (int k = 0; k < BN; ++k) {
            if (sl[k] == li) continue;        // label(k)!=label(i) => k!=i, k!=j
            float tv = sdm[i * BN + k] - dij;
            if (tv < thr) { sum += (double)fmaxf(tv, 0.0f); ++cnt; }
        }
    }
    ss[t] = sum; sc[t] = cnt;
    __syncthreads();
    for (int s = 128; s > 0; s >>= 1) {
        if (t < s) { ss[t] += ss[t+s]; sc[t] += sc[t+s]; }
        __syncthreads();
    }
    if (t == 0) { tpart[blockIdx.x] = ss[0]; tcnt[blockIdx.x] = sc[0]; }
}

// ---------------- quad: compacted positive-pair list, fixed-order build -----
// quad (i,j,k,l) = (pos pair (i,j)) x (neg pair (k,l)) with label(j)!=label(k).
// ~1k positive pairs typical -> 8.7x fewer inner iterations than raw (i,j) scan.
__global__ void __launch_bounds__(256)
k_quad(const float* __restrict__ DM, const int* __restrict__ lbl,
       const float* __restrict__ thrp,
       double* __restrict__ qpart, unsigned* __restrict__ qcnt) {
    __shared__ float pd[NPMAX];               // d_ij of positive pairs
    __shared__ int   pl[NPMAX];               // label of positive pairs
    __shared__ unsigned rowoff[BN + 1];
    __shared__ int sl[BN];
    __shared__ double ss[256];
    __shared__ unsigned sc[256];
    int t = threadIdx.x;
    if (t < BN) sl[t] = lbl[t];
    __syncthreads();
    // deterministic compaction: per-row count -> serial exclusive scan -> write
    if (t < BN) {
        int li = sl[t]; unsigned c = 0;
        for (int j = 0; j < BN; ++j) c += (unsigned)(j != t && sl[j] == li);
        rowoff[t + 1] = c;
    }
    __syncthreads();
    if (t == 0) {
        rowoff[0] = 0;
        for (int r = 1; r <= BN; ++r) rowoff[r] += rowoff[r - 1];
    }
    __syncthreads();
    unsigned npos = rowoff[BN];
    if (t < BN) {
        int li = sl[t]; unsigned off = rowoff[t];
        for (int j = 0; j < BN; ++j)
            if (j != t && sl[j] == li) { pd[off] = DM[t * BN + j]; pl[off] = li; ++off; }
    }
    __syncthreads();
    float thr2 = 0.5f * (*thrp);
    int gid = blockIdx.x * 256 + t;
    int k = gid / BN, l = gid % BN;
    double sum = 0.0; unsigned cnt = 0;
    if (sl[k] != sl[l]) {                     // negative (k,l); implies k!=l
        int lk = sl[k];
        float dkl = DM[k * BN + l];
        for (unsigned p = 0; p < npos; ++p) {
            if (pl[p] == lk) continue;        // label(j)=label(i) != label(k)
            float qv = dkl - pd[p];
            if (qv < thr2) { sum += (double)fmaxf(qv, 0.0f); ++cnt; }
        }
    }
    ss[t] = sum; sc[t] = cnt;
    __syncthreads();
    for (int s = 128; s > 0; s >>= 1) {
        if (t < s) { ss[t] += ss[t+s]; sc[t] += sc[t+s]; }
        __syncthreads();
    }
    if (t == 0) { qpart[blockIdx.x] = ss[0]; qcnt[blockIdx.x] = sc[0]; }
}

// ---------------- fixed-order combine (bit-deterministic) -------------------
__global__ void k_final(const double* __restrict__ TP, const unsigned* __restrict__ TC,
                        const double* __restrict__ QP, const unsigned* __restrict__ QC,
                        float* __restrict__ out) {
    if (threadIdx.x == 0) {
        double ts = 0.0, qs = 0.0;
        unsigned long long tc = 0, qc = 0;
        for (int b = 0; b < NBLK; ++b) {
            ts += TP[b]; tc += TC[b];
            qs += QP[b]; qc += QC[b];
        }
        out[0] = (float)(ts / (double)tc + qs / (double)qc);
    }
}

extern "C" void kernel_launch(void* const* d_in, const int* in_sizes, int n_in,
                              void* d_out, int out_size, void* d_ws, size_t ws_size,
                              hipStream_t stream) {
    const float* E  = (const float*)d_in[0];   // embeddings [96,128] f32
    const int* lbl  = (const int*)d_in[1];     // labels [96] int
    float* out = (float*)d_out;

    // Workspace layout (~38.2 KB); every slot is fully rewritten each call,
    // so no init kernel and no cross-call state.
    float* DM  = (float*)d_ws;                 // [9216] distance matrix
    float* SQ  = DM + BN * BN;                 // [96] row squared norms
    double* TP = (double*)(SQ + BN);           // [36] triplet partial sums (byte 37248, 8-aligned)
    double* QP = TP + NBLK;                    // [36] quad partial sums
    float* THR = (float*)(QP + NBLK);          // [1] threshold
    unsigned* TC = (unsigned*)(THR + 1);       // [36] triplet partial counts
    unsigned* QC = TC + NBLK;                  // [36] quad partial counts

    k_sq     <<<1, 128, 0, stream>>>(E, SQ);
    k_gram_dm<<<NTILE * NTILE, 32, 0, stream>>>(E, SQ, DM);
    k_stats  <<<1, 256, 0, stream>>>(DM, lbl, THR);
    k_triplet<<<NBLK, 256, 0, stream>>>(DM, lbl, THR, TP, TC);
    k_quad   <<<NBLK, 256, 0, stream>>>(DM, lbl, THR, QP, QC);
    k_final  <<<1, 32, 0, stream>>>(TP, TC, QP, QC, out);
}